// NodeVarGraphConvolutionLayer_12249246728260
// MI455X (gfx1250) — compile-verified
//
#include <hip/hip_runtime.h>
#include <hip/hip_bf16.h>

#define NB 16
#define NN 4096
#define NC 64
#define BN (NB * NN)   // 65536 (b,n) pairs

typedef float v2f __attribute__((ext_vector_type(2)));
typedef float v8f __attribute__((ext_vector_type(8)));

// ---------------------------------------------------------------------------
// S[b,n] = sum_c X[b,n,c]; one wave32 per (b,n) row, float2 per lane.
// ---------------------------------------------------------------------------
__global__ __launch_bounds__(256) void rowsum_kernel(const float* __restrict__ X,
                                                     float* __restrict__ S) {
    const int gw   = blockIdx.x * 8 + (threadIdx.x >> 5);   // (b*N+n) row id
    const int lane = threadIdx.x & 31;
    const float2 v = reinterpret_cast<const float2*>(X + (size_t)gw * NC)[lane];
    float s = v.x + v.y;
    #pragma unroll
    for (int m = 16; m; m >>= 1) s += __shfl_xor(s, m, 32);
    if (lane == 0) S[gw] = s;
}

// ---------------------------------------------------------------------------
// pout[b,:] = A[b] @ pin[b,:]  via V_WMMA_F32_16X16X4_F32.
//   A-operand = p[k..k+3] broadcast to all 16 rows (from LDS, 2 VGPRs)
//   B-operand = 16 matrix rows x 4 k's (global_load_b64 per lane, NT hint)
//   C[m,n]    = partial y[r0+n], replicated over m; accumulate over k.
// Each wave owns 16 output rows; block = 8 waves = 128 rows; grid = 32 x B.
// ---------------------------------------------------------------------------
__global__ __launch_bounds__(256) void matvec_wmma_kernel(const float* __restrict__ A,
                                                          const float* __restrict__ pin,
                                                          float* __restrict__ pout) {
    __shared__ float sp[NN];                       // 16 KB: the whole vector p
    const int b   = blockIdx.y;
    const int tid = threadIdx.x;

    // Cooperative load of p into LDS (float4, coalesced).
    const float4* psrc = reinterpret_cast<const float4*>(pin + (size_t)b * NN);
    float4*       pdst = reinterpret_cast<float4*>(sp);
    for (int i = tid; i < NN / 4; i += 256) pdst[i] = psrc[i];
    __syncthreads();

    const int lane = tid & 31;
    const int wave = tid >> 5;
    const int r0   = blockIdx.x * 128 + wave * 16; // first output row of this wave
    const int row  = r0 + (lane & 15);             // matrix row this lane streams
    const int koff = (lane >> 4) << 1;             // 0 for lanes 0-15, 2 for 16-31
    const float* __restrict__ arow = A + ((size_t)b * NN + row) * NN;

    v8f acc = {};
    for (int k0 = 0; k0 < NN; k0 += 8) {
        // B-operand tiles: 2 consecutive f32 of this lane's matrix row (b64 loads, NT)
        v2f bt0 = __builtin_nontemporal_load(
            reinterpret_cast<const v2f*>(arow + k0 + koff));
        v2f bt1 = __builtin_nontemporal_load(
            reinterpret_cast<const v2f*>(arow + k0 + 4 + koff));
        // A-operand tiles: broadcast p chunk from LDS (same 2 addrs per half-wave)
        v2f at0 = *reinterpret_cast<const v2f*>(sp + k0 + koff);
        v2f at1 = *reinterpret_cast<const v2f*>(sp + k0 + 4 + koff);

        acc = __builtin_amdgcn_wmma_f32_16x16x4_f32(false, at0, false, bt0,
                                                    (short)0, acc, false, false);
        acc = __builtin_amdgcn_wmma_f32_16x16x4_f32(false, at1, false, bt1,
                                                    (short)0, acc, false, false);
    }

    // acc VGPR0, lanes 0-15 hold C[M=0, N=lane] = y[r0+lane]
    if (lane < 16) pout[(size_t)b * NN + r0 + lane] = acc[0];
}

// ---------------------------------------------------------------------------
// Y[b,n,c] = sum_i h[i,c,n] * P[i,b,n]; LayerNorm over c; tanh.
// One wave32 per (b,n); each lane owns channels 2*lane, 2*lane+1.
// h (4 MB) stays resident in the 192 MB L2, so the strided gathers are cheap.
// ---------------------------------------------------------------------------
__global__ __launch_bounds__(256) void epilogue_kernel(const float* __restrict__ h,
                                                       const float* __restrict__ P,
                                                       const float* __restrict__ gamma,
                                                       const float* __restrict__ beta,
                                                       float* __restrict__ out) {
    const int gw   = blockIdx.x * 8 + (threadIdx.x >> 5);   // b*N+n
    const int lane = threadIdx.x & 31;
    const int n    = gw & (NN - 1);
    const int c0   = lane * 2;

    float y0 = 0.0f, y1 = 0.0f;
    #pragma unroll
    for (int i = 0; i < 4; ++i) {
        const float pi = P[(size_t)i * BN + gw];            // wave-uniform load
        y0 += h[((size_t)(i * NC + c0)     << 12) + n] * pi;
        y1 += h[((size_t)(i * NC + c0 + 1) << 12) + n] * pi;
    }

    float s  = y0 + y1;
    float s2 = y0 * y0 + y1 * y1;
    #pragma unroll
    for (int m = 16; m; m >>= 1) {
        s  += __shfl_xor(s,  m, 32);
        s2 += __shfl_xor(s2, m, 32);
    }
    const float mu  = s * (1.0f / NC);
    const float var = s2 * (1.0f / NC) - mu * mu;
    const float inv = rsqrtf(var + 1e-5f);

    const float o0 = tanhf((y0 - mu) * inv * gamma[c0]     + beta[c0]);
    const float o1 = tanhf((y1 - mu) * inv * gamma[c0 + 1] + beta[c0 + 1]);
    reinterpret_cast<float2*>(out + ((size_t)gw << 6))[lane] = make_float2(o0, o1);
}

// ---------------------------------------------------------------------------
extern "C" void kernel_launch(void* const* d_in, const int* in_sizes, int n_in,
                              void* d_out, int out_size, void* d_ws, size_t ws_size,
                              hipStream_t stream) {
    const float* A     = (const float*)d_in[0];
    const float* X     = (const float*)d_in[1];
    const float* h     = (const float*)d_in[2];
    const float* gamma = (const float*)d_in[3];
    const float* beta  = (const float*)d_in[4];
    float*       out   = (float*)d_out;

    float* P = (float*)d_ws;                    // P[4][B*N] = 1 MB scratch

    // S = X.sum(-1)  -> P[0]
    rowsum_kernel<<<BN / 8, 256, 0, stream>>>(X, P);

    // P[i] = A @ P[i-1], i = 1..3  (memory-bound WMMA GEMV, full-grid launches)
    dim3 grid(NN / 128, NB);
    matvec_wmma_kernel<<<grid, 256, 0, stream>>>(A, P,            P + BN);
    matvec_wmma_kernel<<<grid, 256, 0, stream>>>(A, P + BN,       P + 2 * BN);
    matvec_wmma_kernel<<<grid, 256, 0, stream>>>(A, P + 2 * BN,   P + 3 * BN);

    // channel mix + LayerNorm + tanh
    epilogue_kernel<<<BN / 8, 256, 0, stream>>>(h, P, gamma, beta, out);
}